// Emu3VQVAEVectorQuantizer_26611617366714
// MI455X (gfx1250) — compile-verified
//
#include <hip/hip_runtime.h>
#include <hip/hip_bf16.h>
#include <stdint.h>

// VQ nearest-codebook-index search (Emu3 VQVAE quantizer), CDNA5 / gfx1250.
//   argmin_k ||x - e_k||^2  ==  argmin_k ( ||e_k||^2 - 2 x.e_k )
// One V_WMMA_F32_16X16X4_F32 (K=4 == EMBED_DIM) produces a 16(rows)x16(codes)
// score tile: A = -2*x (16x4), B = e^T (4x16), C = per-column ||e||^2.
// Codebook tiles are staged global->LDS with GLOBAL_LOAD_ASYNC_TO_LDS_B128
// (ASYNCcnt, double-buffered); row norms are computed ONCE per block per chunk
// into LDS so the 8-way redundant per-lane norm math disappears from the loop.

typedef float v2f __attribute__((ext_vector_type(2)));
typedef float v8f __attribute__((ext_vector_type(8)));

#define CODEBOOK 32768
#define EMBED    4
#define NROWS    9216      // 1*4*48*48
#define TPER     2304      // 48*48 (spatial elements per (t,c) plane)
#define CHUNK    128       // codebook rows staged per block per step (2 KB)
#define NCHUNK   (CODEBOOK / CHUNK)   // 256

__global__ __launch_bounds__(128)
void vq_argmin_wmma_kernel(const float* __restrict__ hidden,  // [1,4,4,48,48] f32
                           const float* __restrict__ emb,     // [32768,4]     f32
                           int* __restrict__ out)             // [9216]        i32
{
    __shared__ float4 lbuf[2][CHUNK];   // double-buffered codebook rows, 2 x 2KB
    __shared__ float  lnorm[2][CHUNK];  // per-row ||e||^2, computed once per block

    const int tid     = threadIdx.x;
    const int lane    = tid & 31;
    const int wave    = blockIdx.x * 4 + (tid >> 5);
    const int rowBase = wave * 16;      // grid sized exactly: 576 waves x 16 rows

    // ---- A tile: -2 * x[rowBase + (lane&15)][k0 .. k0+1], k0 = (lane>=16 ? 2 : 0)
    // ISA 7.12.2 "32-bit A-Matrix 16x4": lanes 0-15 -> K=0,1; lanes 16-31 -> K=2,3.
    const int m   = lane & 15;
    const int kk  = (lane >> 4) * 2;
    const int row = rowBase + m;
    const int t   = row / TPER;
    const int r   = row - t * TPER;
    v2f a;  // x[row][c] = hidden[t*(EMBED*TPER) + c*TPER + r] (channels-first layout)
    a.x = -2.0f * hidden[t * (EMBED * TPER) + (kk + 0) * TPER + r];
    a.y = -2.0f * hidden[t * (EMBED * TPER) + (kk + 1) * TPER + r];

    v8f best;
    int bestIdx[8];
#pragma unroll
    for (int i = 0; i < 8; ++i) { best[i] = 3.4e38f; bestIdx[i] = 0; }

    // Async stage of one 128-row codebook chunk: each of the 128 threads DMAs
    // one float4 row straight into LDS (no VGPR round-trip), ASYNCcnt-tracked.
    auto issue_chunk = [&](int c, int buf) {
        const float4* gsrc = (const float4*)emb + (size_t)c * CHUNK + tid;
        uint32_t ldsAddr = (uint32_t)(uintptr_t)(&lbuf[buf][tid]); // low 32b of flat shared ptr = LDS offset
        uint64_t gaddr   = (uint64_t)(uintptr_t)gsrc;
        asm volatile("global_load_async_to_lds_b128 %0, %1, off"
                     :: "v"(ldsAddr), "v"(gaddr) : "memory");
    };

    issue_chunk(0, 0);

    for (int c = 0; c < NCHUNK; ++c) {
        const int cb = c & 1;
        asm volatile("s_wait_asynccnt 0x0" ::: "memory");  // my DMA for buf cb done
        __syncthreads();                                   // all DMA landed; prev reads drained
        if (c + 1 < NCHUNK) issue_chunk(c + 1, cb ^ 1);    // overlap next stage

        // Each thread computes ONE staged row's norm (vs 8x redundant per-lane math).
        {
            const float4 v = lbuf[cb][tid];
            lnorm[cb][tid] = v.x * v.x + v.y * v.y + v.z * v.z + v.w * v.w;
        }
        __syncthreads();                                   // norms visible block-wide

        const float* bufF = (const float*)lbuf[cb];
        const float* nrmF = lnorm[cb];
        const int    cBase = c * CHUNK;
#pragma unroll
        for (int tb = 0; tb < CHUNK / 16; ++tb) {
            const int ci  = tb * 16 + m;                   // row within chunk
            const int col = cBase + ci;                    // this lane's codebook column

            // B tile (4x16): lanes 0-15 need e[0..1], lanes 16-31 need e[2..3]
            // -> direct ds_load_b64 at kk-float offset, no b128 + selects.
            const v2f  b   = *(const v2f*)(bufF + ci * 4 + kk);
            const float nrm = nrmF[ci];                    // ds_load_b32, shared norm

            // C bias ||e_n||^2: column N depends only on lane&15 in the C/D layout,
            // so a per-lane splat across the 8 accumulator VGPRs is the broadcast.
            v8f cc;
#pragma unroll
            for (int i = 0; i < 8; ++i) cc[i] = nrm;

            v8f d = __builtin_amdgcn_wmma_f32_16x16x4_f32(
                false, a, false, b, (short)0, cc, false, false);

#pragma unroll
            for (int i = 0; i < 8; ++i) {
                if (d[i] < best[i]) { best[i] = d[i]; bestIdx[i] = col; }
            }
        }
    }

    // ---- Branchless argmin reduce across the 16 columns held by lanes.
    // Monotonic float->u32 map, pack (score,index) into u64, min-reduce.
    // XOR masks 1,2,4,8 stay inside each 16-lane half of the wave32, matching
    // the C/D layout split (lanes 0-15: M=0..7, lanes 16-31: M=8..15).
    unsigned long long key[8];
#pragma unroll
    for (int i = 0; i < 8; ++i) {
        uint32_t u = __float_as_uint(best[i]);
        u ^= (uint32_t)((int32_t)u >> 31) | 0x80000000u;   // order-preserving map
        key[i] = ((unsigned long long)u << 32) | (uint32_t)bestIdx[i];
    }
#pragma unroll
    for (int mask = 1; mask < 16; mask <<= 1) {
#pragma unroll
        for (int i = 0; i < 8; ++i) {
            unsigned long long o = __shfl_xor(key[i], mask, 32);
            key[i] = (o < key[i]) ? o : key[i];            // exact tie -> lower index
        }
    }

    if ((lane & 15) == 0) {
        const int mOff = (lane >> 4) * 8;   // lane 0 -> rows 0..7, lane 16 -> rows 8..15
#pragma unroll
        for (int i = 0; i < 8; ++i) out[rowBase + mOff + i] = (int)(uint32_t)key[i];
    }
}

extern "C" void kernel_launch(void* const* d_in, const int* in_sizes, int n_in,
                              void* d_out, int out_size, void* d_ws, size_t ws_size,
                              hipStream_t stream) {
    const float* hidden = (const float*)d_in[0];   // 36864 f32
    const float* emb    = (const float*)d_in[1];   // 131072 f32
    int* out = (int*)d_out;                        // 9216 int32 indices

    dim3 block(128);                 // 4 waves -> 64 WMMA rows per block
    dim3 grid(NROWS / 16 / 4);       // 144 blocks, exact cover (no tail)
    hipLaunchKernelGGL(vq_argmin_wmma_kernel, grid, block, 0, stream, hidden, emb, out);
}